// GNNMLPLayer_6236292513986
// MI455X (gfx1250) — compile-verified
//
#include <hip/hip_runtime.h>
#include <math.h>

#define N_NODES 50000
#define N_EDGES 1600000
#define DD 128
#define GG 128

typedef float v2f __attribute__((ext_vector_type(2)));
typedef float v8f __attribute__((ext_vector_type(8)));

// ---------- monotone ordered key for float atomic-max over uint ----------
__device__ __forceinline__ unsigned f2key(float f) {
    unsigned u = __float_as_uint(f);
    return (u & 0x80000000u) ? ~u : (u | 0x80000000u);
}
__device__ __forceinline__ float key2f(unsigned k) {
    unsigned u = (k & 0x80000000u) ? (k ^ 0x80000000u) : ~k;
    return __uint_as_float(u);
}

// ---------- 1) per-node dots with we_dst / we_src (one wave per node) ----------
__global__ void k_node_dots(const float* __restrict__ nf,
                            const float* __restrict__ wd,
                            const float* __restrict__ wsrc,
                            float* __restrict__ ld, float* __restrict__ ls) {
    int gid  = blockIdx.x * blockDim.x + threadIdx.x;
    int node = gid >> 5;
    int lane = gid & 31;
    if (node >= N_NODES) return;
    float4 v = ((const float4*)(nf + (size_t)node * DD))[lane];
    float4 a = ((const float4*)wd)[lane];
    float4 b = ((const float4*)wsrc)[lane];
    float sd = v.x*a.x + v.y*a.y + v.z*a.z + v.w*a.w;
    float ss = v.x*b.x + v.y*b.y + v.z*b.z + v.w*b.w;
    for (int off = 16; off; off >>= 1) {
        sd += __shfl_down(sd, off);
        ss += __shfl_down(ss, off);
    }
    if (lane == 0) { ld[node] = sd; ls[node] = ss; }
}

// ---------- 2) edge logits + LeakyReLU + segment max ----------
__global__ void k_edge_logit_max(const int* __restrict__ src, const int* __restrict__ dst,
                                 const float* __restrict__ ld, const float* __restrict__ ls,
                                 const float* __restrict__ be,
                                 float* __restrict__ logit, unsigned* __restrict__ maxkey) {
    int e = blockIdx.x * blockDim.x + threadIdx.x;
    if (e >= N_EDGES) return;
    int d = dst[e];
    float l = ld[d] + ls[src[e]] + be[0];
    l = (l >= 0.0f) ? l : 0.01f * l;
    logit[e] = l;
    atomicMax(&maxkey[d], f2key(l));
}

// ---------- 3) exp(logit - max) in place + segment sum ----------
__global__ void k_edge_exp_sum(const int* __restrict__ dst,
                               const unsigned* __restrict__ maxkey,
                               float* __restrict__ logit_ev, float* __restrict__ denom) {
    int e = blockIdx.x * blockDim.x + threadIdx.x;
    if (e >= N_EDGES) return;
    int d = dst[e];
    float ev = __expf(logit_ev[e] - key2f(maxkey[d]));
    logit_ev[e] = ev;
    unsafeAtomicAdd(&denom[d], ev);
}

// ---------- 4) generic fp32 WMMA GEMM: Out[M,128] = act(A[M,K] @ W[K,128] + b) ----------
// MODE 0: A = A0 (plain), no activation
// MODE 1: A = [ elu(A0) | A1 ] (K must be 256), relu output
// MODE 2: A = A0 (plain), relu output
template <int K, int MODE>
__global__ __launch_bounds__(256)
void k_gemm(const float* __restrict__ A0, const float* __restrict__ A1,
            const float* __restrict__ W, const float* __restrict__ bias,
            float* __restrict__ Out, int M) {
    __shared__ float As[64][68];     // K-chunk contiguous per row; 272 B row stride
    __shared__ float Wst[128][66];   // TRANSPOSED: [col][k]; 66*4=264 B, col*66 even -> b64 aligned

    const int tid  = threadIdx.x;
    const int wave = tid >> 5;
    const int lane = tid & 31;
    const int rowBase = blockIdx.x * 64;
    const int m0    = (wave >> 1) * 16;   // local row tile base (0/16/32/48)
    const int nBase = (wave & 1) * 64;    // this wave owns 4 col tiles of 16
    const int lane15 = lane & 15;
    const int khalf  = (lane >> 4) << 1;  // lanes 0-15 -> K+0/1, lanes 16-31 -> K+2/3

    v8f acc[4];
#pragma unroll
    for (int t = 0; t < 4; ++t)
#pragma unroll
        for (int i = 0; i < 8; ++i) acc[t][i] = 0.0f;

    for (int k0 = 0; k0 < K; k0 += 64) {
        // ---- stage A tile: 64 rows x 64 cols = 1024 float4, 4 per thread ----
#pragma unroll
        for (int i = 0; i < 4; ++i) {
            int idx = tid + i * 256;
            int r   = idx >> 4;
            int c4  = (idx & 15) * 4;
            int grow = rowBase + r;
            float4 v = make_float4(0.f, 0.f, 0.f, 0.f);
            if (grow < M) {
                if (MODE == 1) {
                    if (k0 < 128) {
                        v = *(const float4*)(A0 + (size_t)grow * 128 + k0 + c4);
                        v.x = v.x > 0.f ? v.x : __expf(v.x) - 1.f;
                        v.y = v.y > 0.f ? v.y : __expf(v.y) - 1.f;
                        v.z = v.z > 0.f ? v.z : __expf(v.z) - 1.f;
                        v.w = v.w > 0.f ? v.w : __expf(v.w) - 1.f;
                    } else {
                        v = *(const float4*)(A1 + (size_t)grow * 128 + (k0 - 128) + c4);
                    }
                } else {
                    v = *(const float4*)(A0 + (size_t)grow * K + k0 + c4);
                }
            }
            *(float4*)&As[r][c4] = v;
        }
        // ---- stage W tile TRANSPOSED: coalesced float4 global read, scalar DS scatter ----
#pragma unroll
        for (int i = 0; i < 8; ++i) {
            int idx = tid + i * 256;       // 0..2047
            int r   = idx >> 5;            // k within chunk: 0..63
            int c4  = (idx & 31) * 4;      // col: 0..124
            float4 v = *(const float4*)(W + (size_t)(k0 + r) * 128 + c4);
            Wst[c4 + 0][r] = v.x;
            Wst[c4 + 1][r] = v.y;
            Wst[c4 + 2][r] = v.z;
            Wst[c4 + 3][r] = v.w;
        }
        __syncthreads();

        // ---- compute: software-pipelined fragment double-buffer ----
        const float* arow = &As[m0 + lane15][0];
        v2f a_cur = *(const v2f*)(arow + khalf);
        v2f b_cur[4];
#pragma unroll
        for (int t = 0; t < 4; ++t)
            b_cur[t] = *(const v2f*)(&Wst[nBase + t * 16 + lane15][0] + khalf);

#pragma unroll
        for (int kk = 0; kk < 64; kk += 4) {
            v2f a_nxt = a_cur;
            v2f b_nxt[4];
#pragma unroll
            for (int t = 0; t < 4; ++t) b_nxt[t] = b_cur[t];
            if (kk + 4 < 64) {
                a_nxt = *(const v2f*)(arow + kk + 4 + khalf);
#pragma unroll
                for (int t = 0; t < 4; ++t)
                    b_nxt[t] = *(const v2f*)(&Wst[nBase + t * 16 + lane15][0] + kk + 4 + khalf);
            }
#pragma unroll
            for (int t = 0; t < 4; ++t)
                acc[t] = __builtin_amdgcn_wmma_f32_16x16x4_f32(
                    false, a_cur, false, b_cur[t], (short)0, acc[t], false, false);
            a_cur = a_nxt;
#pragma unroll
            for (int t = 0; t < 4; ++t) b_cur[t] = b_nxt[t];
        }
        __syncthreads();
    }

    // writeback: C/D layout -> lanes 0-15: M = m0+i, lanes 16-31: M = m0+i+8; col = lane&15
#pragma unroll
    for (int t = 0; t < 4; ++t) {
        const int col = nBase + t * 16 + lane15;
        const float bv = bias[col];
#pragma unroll
        for (int i = 0; i < 8; ++i) {
            int grow = rowBase + m0 + i + ((lane >> 4) << 3);
            if (grow < M) {
                float v = acc[t][i] + bv;
                if (MODE != 0) v = v > 0.f ? v : 0.f;
                Out[(size_t)grow * 128 + col] = v;
            }
        }
    }
}

// ---------- 5) edge aggregation: c[dst] += hv[src] * a  (one wave per edge) ----------
__global__ void k_aggregate(const int* __restrict__ src, const int* __restrict__ dst,
                            const float* __restrict__ ev, const float* __restrict__ denom,
                            const float* __restrict__ hv, float* __restrict__ c) {
    long gid = (long)blockIdx.x * blockDim.x + threadIdx.x;
    long e   = gid >> 5;
    int lane = (int)(gid & 31);
    if (e >= N_EDGES) return;
    int d = dst[e], s = src[e];
    float a = ev[e] / denom[d];
    float4 v = ((const float4*)(hv + (size_t)s * GG))[lane];
    float* cp = c + (size_t)d * GG + lane * 4;
    unsafeAtomicAdd(cp + 0, v.x * a);
    unsafeAtomicAdd(cp + 1, v.y * a);
    unsafeAtomicAdd(cp + 2, v.z * a);
    unsafeAtomicAdd(cp + 3, v.w * a);
}

// ---------- 6) column sums / sumsq for BatchNorm ----------
__global__ void k_stats(const float* __restrict__ h, float* __restrict__ sums) {
    int j = threadIdx.x;  // 0..127
    float s = 0.f, q = 0.f;
    for (int r = blockIdx.x; r < N_NODES; r += gridDim.x) {
        float v = h[(size_t)r * 128 + j];
        s += v; q += v * v;
    }
    unsafeAtomicAdd(&sums[j], s);
    unsafeAtomicAdd(&sums[128 + j], q);
}

// ---------- 7) BatchNorm normalize ----------
__global__ void k_norm(const float* __restrict__ h, const float* __restrict__ sums,
                       const float* __restrict__ gamma, const float* __restrict__ beta,
                       float* __restrict__ out) {
    long idx = (long)blockIdx.x * blockDim.x + threadIdx.x;
    if (idx >= (long)N_NODES * 128) return;
    int j = (int)(idx & 127);
    const float invN = 1.0f / (float)N_NODES;
    float mean = sums[j] * invN;
    float var  = sums[128 + j] * invN - mean * mean;
    float v = h[idx];
    out[idx] = gamma[j] * (v - mean) * rsqrtf(var + 1e-5f) + beta[j];
}

extern "C" void kernel_launch(void* const* d_in, const int* in_sizes, int n_in,
                              void* d_out, int out_size, void* d_ws, size_t ws_size,
                              hipStream_t stream) {
    (void)in_sizes; (void)n_in; (void)out_size; (void)ws_size;
    const float* nf     = (const float*)d_in[0];
    const int*   src    = (const int*)  d_in[1];
    const int*   dst    = (const int*)  d_in[2];
    const float* we_dst = (const float*)d_in[3];
    const float* we_src = (const float*)d_in[4];
    const float* be     = (const float*)d_in[5];
    const float* w_proj = (const float*)d_in[6];
    const float* b_proj = (const float*)d_in[7];
    const float* w1     = (const float*)d_in[8];
    const float* b1     = (const float*)d_in[9];
    const float* w2     = (const float*)d_in[10];
    const float* b2     = (const float*)d_in[11];
    const float* gamma  = (const float*)d_in[12];
    const float* beta   = (const float*)d_in[13];
    float* out = (float*)d_out;

    // workspace carve-up (256B aligned)
    char* ws = (char*)d_ws;
    size_t off = 0;
    auto carve = [&](size_t bytes) -> void* {
        void* p = ws + off;
        off = (off + bytes + 255) & ~(size_t)255;
        return p;
    };
    float*    ld     = (float*)   carve((size_t)N_NODES * 4);
    float*    ls     = (float*)   carve((size_t)N_NODES * 4);
    unsigned* maxkey = (unsigned*)carve((size_t)N_NODES * 4);
    float*    denom  = (float*)   carve((size_t)N_NODES * 4);
    float*    ev     = (float*)   carve((size_t)N_EDGES * 4);           // logit, then ev in place
    float*    hv     = (float*)   carve((size_t)N_NODES * GG * 4);      // later reused as h1
    float*    cbuf   = (float*)   carve((size_t)N_NODES * GG * 4);      // later reused as h2
    float*    stats  = (float*)   carve(256 * 4);
    float* h1 = hv;    // hv dead after aggregation
    float* h2 = cbuf;  // c dead after GEMM1

    // zero accumulators (graph-capture-safe)
    hipMemsetAsync(maxkey, 0, (size_t)N_NODES * 4, stream);
    hipMemsetAsync(denom,  0, (size_t)N_NODES * 4, stream);
    hipMemsetAsync(cbuf,   0, (size_t)N_NODES * GG * 4, stream);
    hipMemsetAsync(stats,  0, 256 * 4, stream);

    const int nGemmBlocks = (N_NODES + 63) / 64;  // 782

    k_node_dots<<<(N_NODES * 32 + 255) / 256, 256, 0, stream>>>(nf, we_dst, we_src, ld, ls);
    k_edge_logit_max<<<(N_EDGES + 255) / 256, 256, 0, stream>>>(src, dst, ld, ls, be, ev, maxkey);
    k_edge_exp_sum<<<(N_EDGES + 255) / 256, 256, 0, stream>>>(dst, maxkey, ev, denom);
    k_gemm<128, 0><<<nGemmBlocks, 256, 0, stream>>>(nf, nullptr, w_proj, b_proj, hv, N_NODES);
    k_aggregate<<<(int)(((long)N_EDGES * 32 + 255) / 256), 256, 0, stream>>>(src, dst, ev, denom, hv, cbuf);
    k_gemm<256, 1><<<nGemmBlocks, 256, 0, stream>>>(cbuf, nf, w1, b1, h1, N_NODES);
    k_gemm<128, 2><<<nGemmBlocks, 256, 0, stream>>>(h1, nullptr, w2, b2, h2, N_NODES);
    k_stats<<<256, 128, 0, stream>>>(h2, stats);
    k_norm<<<(int)(((long)N_NODES * 128 + 255) / 256), 256, 0, stream>>>(h2, stats, gamma, beta, out);
}